// ProteinProteinInteractionPrediction_91053306675577
// MI455X (gfx1250) — compile-verified
//
#include <hip/hip_runtime.h>
#include <math.h>

#define S 32          // padded row stride in floats
#define DIMN 20
#define HH 10
#define G3 30         // 3*H
#define NL 3
#define ESMN 1280
#define KS 4          // split-K factor for adjacency GEMM

typedef __attribute__((ext_vector_type(2))) float v2f;
typedef __attribute__((ext_vector_type(8))) float v8f;

// ---------------- ESM linear + relu : e[p][d] = relu(wesm[d,:] . feat_p + b[d]) ----
__global__ void __launch_bounds__(256)
k_esm(const float* __restrict__ f1, const float* __restrict__ f2,
      const float* __restrict__ w, const float* __restrict__ b,
      float* __restrict__ e1, float* __restrict__ e2) {
  __shared__ float red[256];
  int p = blockIdx.x / DIMN;
  int d = blockIdx.x % DIMN;
  const float* f = p ? f2 : f1;
  float acc = 0.f;
  for (int e = threadIdx.x; e < ESMN; e += 256)
    acc = fmaf(f[e], w[d * ESMN + e], acc);
  red[threadIdx.x] = acc; __syncthreads();
  for (int s = 128; s > 0; s >>= 1) {
    if (threadIdx.x < s) red[threadIdx.x] += red[threadIdx.x + s];
    __syncthreads();
  }
  if (threadIdx.x == 0) {
    float v = red[0] + b[d];
    (p ? e2 : e1)[d] = v > 0.f ? v : 0.f;
  }
}

// ---------------- initial linear (both proteins via blockIdx.y) -----------------
__global__ void __launch_bounds__(256)
k_init(const float* __restrict__ p1, const float* __restrict__ p2,
       const float* __restrict__ w1, const float* __restrict__ b1,
       const float* __restrict__ w2, const float* __restrict__ b2,
       float* __restrict__ x1, float* __restrict__ x2, int n) {
  const int p = blockIdx.y;
  const float* prot = p ? p2 : p1;
  const float* w = p ? w2 : w1;
  const float* b = p ? b2 : b1;
  float* xs = p ? x2 : x1;
  int idx = blockIdx.x * 256 + threadIdx.x;
  if (idx >= n * S) return;
  int c = idx & (S - 1), r = idx >> 5;
  float v = 0.f;
  if (c < DIMN) {
    v = b[c];
    #pragma unroll
    for (int k = 0; k < 7; ++k) v = fmaf(prot[r * 7 + k], w[c * 7 + k], v);
  }
  xs[idx] = v;
}

// ---------------- conv1d k=3 pad=1 over sequence dim (both proteins) ------------
__global__ void __launch_bounds__(256)
k_conv(const float* __restrict__ x1, const float* __restrict__ x2,
       const float* __restrict__ w1, const float* __restrict__ w2,
       const float* __restrict__ b1, const float* __restrict__ b2,
       float* __restrict__ y1, float* __restrict__ y2, int n) {
  const int p = blockIdx.y;
  const float* x = p ? x2 : x1;
  const float* w = p ? w2 : w1;
  const float* b = p ? b2 : b1;
  float* y = p ? y2 : y1;
  int idx = blockIdx.x * 256 + threadIdx.x;
  if (idx >= n * S) return;
  int o = idx & (S - 1), r = idx >> 5;
  float v = 0.f;
  if (o < DIMN) {
    v = b[o];
    #pragma unroll
    for (int k = 0; k < 3; ++k) {
      int rr = r + k - 1;
      if (rr < 0 || rr >= n) continue;
      const float* xr = x + (size_t)rr * S;
      #pragma unroll
      for (int i = 0; i < DIMN; ++i)
        v = fmaf(xr[i], w[o * 60 + i * 3 + k], v);
    }
  }
  y[idx] = v;
}

// ---------------- GRU input gates, permuted layout [t][3k+{r,z,n}] --------------
__global__ void __launch_bounds__(256)
k_gates(const float* __restrict__ x1, const float* __restrict__ x2,
        const float* __restrict__ wi1, const float* __restrict__ wi2,
        const float* __restrict__ bi1, const float* __restrict__ bi2,
        float* __restrict__ g1, float* __restrict__ g2, int n) {
  const int p = blockIdx.y;
  const float* x  = p ? x2 : x1;
  const float* wih = p ? wi2 : wi1;
  const float* bih = p ? bi2 : bi1;
  float* gi = p ? g2 : g1;
  int idx = blockIdx.x * 256 + threadIdx.x;   // 2*n*S threads
  if (idx >= 2 * n * S) return;
  int j = idx & (S - 1);
  int t = (idx >> 5) % n;
  int d = idx / (n * S);
  if (j < G3) {
    int tt = d ? (n - 1 - t) : t;             // backward dir sees reversed sequence
    const float* xr = x + (size_t)tt * S;
    float v = bih[d * G3 + j];
    #pragma unroll
    for (int c = 0; c < DIMN; ++c)
      v = fmaf(xr[c], wih[d * G3 * DIMN + j * DIMN + c], v);
    int g = j / HH, kk = j - g * HH;          // gate group, hidden index
    gi[idx - j + kk * 3 + g] = v;             // permuted: lane k reads 3k..3k+2
  } else {
    gi[idx] = 0.f;
  }
}

// ---------------- serial GRU scan: 4 blocks = {protein} x {direction} -----------
// Lane k (<10) owns Whh rows k, 10+k, 20+k -> gates computed with no cross-lane
// traffic; h broadcast with 10 v_readlane into scalar regs.
__global__ void __launch_bounds__(32)
k_gru_scan(const float* __restrict__ gi1, const float* __restrict__ gi2,
           const float* __restrict__ wh1, const float* __restrict__ wh2,
           const float* __restrict__ bh1, const float* __restrict__ bh2,
           float* __restrict__ o1, float* __restrict__ o2, int n) {
  const int p   = blockIdx.x >> 1;
  const int dir = blockIdx.x & 1;
  const int lane = threadIdx.x;
  const bool act = lane < HH;
  const float* gi  = (p ? gi2 : gi1) + (size_t)dir * n * S;
  const float* whh = (p ? wh2 : wh1) + dir * G3 * HH;
  const float* bhh = (p ? bh2 : bh1) + dir * G3;
  float* out = (p ? o2 : o1) + (dir ? HH : 0);
  float wr[HH], wz[HH], wn[HH];
  float br = 0.f, bz = 0.f, bn = 0.f;
  #pragma unroll
  for (int c = 0; c < HH; ++c) { wr[c] = 0.f; wz[c] = 0.f; wn[c] = 0.f; }
  if (act) {
    #pragma unroll
    for (int c = 0; c < HH; ++c) {
      wr[c] = whh[lane * HH + c];
      wz[c] = whh[(HH + lane) * HH + c];
      wn[c] = whh[(2 * HH + lane) * HH + c];
    }
    br = bhh[lane]; bz = bhh[HH + lane]; bn = bhh[2 * HH + lane];
  }
  float h[HH];
  #pragma unroll
  for (int c = 0; c < HH; ++c) h[c] = 0.f;
  float hk = 0.f;                                   // this lane's own h value
  float gr = 0.f, gz = 0.f, gn = 0.f;
  if (act) { const float* g0 = gi + 3 * lane; gr = g0[0]; gz = g0[1]; gn = g0[2]; }
  for (int t = 0; t < n; ++t) {
    float pr = 0.f, pz = 0.f, pn = 0.f;             // prefetch next step's gates
    if (act && t + 1 < n) {
      const float* gx = gi + (size_t)(t + 1) * S + 3 * lane;
      pr = gx[0]; pz = gx[1]; pn = gx[2];
    }
    float ar = br, a2r = 0.f, az = bz, a2z = 0.f, an = bn, a2n = 0.f;
    #pragma unroll
    for (int c = 0; c < 5; ++c) {                   // two 5-deep chains per gate
      ar  = fmaf(h[c],     wr[c],     ar);
      a2r = fmaf(h[c + 5], wr[c + 5], a2r);
      az  = fmaf(h[c],     wz[c],     az);
      a2z = fmaf(h[c + 5], wz[c + 5], a2z);
      an  = fmaf(h[c],     wn[c],     an);
      a2n = fmaf(h[c + 5], wn[c + 5], a2n);
    }
    float ghr = ar + a2r, ghz = az + a2z, ghn = an + a2n;
    float r = __fdividef(1.f, 1.f + __expf(-(gr + ghr)));
    float z = __fdividef(1.f, 1.f + __expf(-(gz + ghz)));
    float e2 = __expf(2.f * (gn + r * ghn));
    float nv = __fdividef(e2 - 1.f, e2 + 1.f);      // tanh
    float hn = (1.f - z) * nv + z * hk;
    if (act) out[(size_t)(dir ? (n - 1 - t) : t) * S + lane] = hn;
    #pragma unroll
    for (int c = 0; c < HH; ++c)                    // broadcast h -> scalar regs
      h[c] = __uint_as_float(__builtin_amdgcn_readlane(__float_as_uint(hn), c));
    hk = hn;
    gr = pr; gz = pz; gn = pn;
  }
}

// ---------------- wgnn linear + relu (shared weights, both proteins) ------------
__global__ void __launch_bounds__(256)
k_wgnn(const float* __restrict__ x1, const float* __restrict__ x2,
       const float* __restrict__ w, const float* __restrict__ b,
       float* __restrict__ h1, float* __restrict__ h2, int n) {
  const int p = blockIdx.y;
  const float* x = p ? x2 : x1;
  float* hs = p ? h2 : h1;
  int idx = blockIdx.x * 256 + threadIdx.x;
  if (idx >= n * S) return;
  int d = idx & (S - 1), r = idx >> 5;
  float v = 0.f;
  if (d < DIMN) {
    v = b[d];
    const float* xr = x + (size_t)r * S;
    #pragma unroll
    for (int c = 0; c < DIMN; ++c) v = fmaf(xr[c], w[d * DIMN + c], v);
    v = v > 0.f ? v : 0.f;
  }
  hs[idx] = v;
}

// ---------------- adjacency GEMM via V_WMMA_F32_16X16X4_F32, split-K ------------
// part[p][kc][n][32] = A_p[rows, kchunk] @ Hs_p[kchunk, 32]
// A-frag (16x4 f32): lanes 0-15 K=0,1; lanes 16-31 K=2,3.
// B-frag (4x16 f32): lanes 0-15 rows K=0,1; lanes 16-31 rows K=2,3.
// Pointer-increment loop + unroll 8 so the backend clauses loads and
// software-pipelines the v_wmma stream (verified codegen shape in round 1).
__global__ void __launch_bounds__(256)
k_adj_wmma(const float* __restrict__ A1, const float* __restrict__ A2,
           const float* __restrict__ Hs1, const float* __restrict__ Hs2,
           float* __restrict__ part, int n) {
  const int wave = threadIdx.x >> 5;                 // 8 waves / block
  const int lane = threadIdx.x & 31;
  const int kc = blockIdx.y;
  const int p  = blockIdx.z;
  const float* A  = p ? A2 : A1;
  const float* Hs = p ? Hs2 : Hs1;
  const int rowBase = (blockIdx.x * 8 + wave) * 16;
  if (rowBase >= n) return;                          // wave-uniform
  const int kLen = n / KS;
  const int k0 = kc * kLen;
  const int mrow = rowBase + (lane & 15);
  const float* ap = A + (size_t)mrow * n + ((lane >> 4) * 2) + k0;
  const float* hp = Hs + (size_t)(k0 + ((lane >> 4) * 2)) * S + (lane & 15);
  v8f c0 = {}; v8f c1 = {};
  const int kIters = kLen >> 2;                      // 4 K per iteration
  #pragma unroll 8
  for (int it = 0; it < kIters; ++it) {
    v2f a;
    a.x = ap[0];
    a.y = ap[1];
    v2f b0, b1;
    b0.x = hp[0];        b0.y = hp[S];
    b1.x = hp[16];       b1.y = hp[S + 16];
    c0 = __builtin_amdgcn_wmma_f32_16x16x4_f32(false, a, false, b0,
                                               (short)0, c0, false, false);
    c1 = __builtin_amdgcn_wmma_f32_16x16x4_f32(false, a, false, b1,
                                               (short)0, c1, false, false);
    ap += 4;
    hp += 4 * S;
  }
  const int rstore = rowBase + (lane >> 4) * 8;
  float* o = part + ((size_t)(p * KS + kc) * n + rstore) * S + (lane & 15);
  #pragma unroll
  for (int r = 0; r < 8; ++r) {
    o[(size_t)r * S]      = c0[r];
    o[(size_t)r * S + 16] = c1[r];
  }
}

// ---------------- reduce split-K partials into xs -------------------------------
__global__ void __launch_bounds__(256)
k_adj_red(const float* __restrict__ part, float* __restrict__ x1,
          float* __restrict__ x2, int n) {
  int idx = blockIdx.x * 256 + threadIdx.x;
  if (idx >= n * S) return;
  const int p = blockIdx.y;
  const float* pp = part + (size_t)p * KS * n * S;
  float v = pp[idx];
  #pragma unroll
  for (int kc = 1; kc < KS; ++kc) v += pp[(size_t)kc * n * S + idx];
  (p ? x2 : x1)[idx] = v;
}

// ---------------- fusion: h = lam*esm + (1-lam)*xs, zero pads -------------------
__global__ void __launch_bounds__(256)
k_fuse(const float* __restrict__ x1, const float* __restrict__ x2,
       const float* __restrict__ e1, const float* __restrict__ e2,
       const float* __restrict__ plam,
       float* __restrict__ h1, float* __restrict__ h2, int n) {
  const int p = blockIdx.y;
  const float* xs = p ? x2 : x1;
  const float* esm = p ? e2 : e1;
  float* h = p ? h2 : h1;
  int idx = blockIdx.x * 256 + threadIdx.x;
  if (idx >= n * S) return;
  int c = idx & (S - 1);
  float v = 0.f;
  if (c < DIMN) {
    float lam = __fdividef(1.f, 1.f + __expf(-plam[0]));
    v = lam * esm[c] + (1.f - lam) * xs[idx];
  }
  h[idx] = v;
}

// ---------------- attention pass 1: per-row softmax stats (both directions) -----
__global__ void __launch_bounds__(256)
k_attn_stats(const float* __restrict__ h1, const float* __restrict__ h2,
             float* __restrict__ m1, float* __restrict__ l1,
             float* __restrict__ m2, float* __restrict__ l2, int n) {
  __shared__ float hi[DIMN];
  __shared__ float sm[256], sl[256];
  const int p = blockIdx.y;
  const float* hA = p ? h2 : h1;
  const float* hB = p ? h1 : h2;
  float* mOut = p ? m2 : m1;
  float* lOut = p ? l2 : l1;
  const int i = blockIdx.x;
  if (threadIdx.x < DIMN) hi[threadIdx.x] = hA[(size_t)i * S + threadIdx.x];
  __syncthreads();
  const float scale = 0.22360679774997896f;   // 1/sqrt(20)
  float m = -3.0e38f, l = 0.f;
  for (int j = threadIdx.x; j < n; j += 256) {
    const float* hb = hB + (size_t)j * S;
    float sdot = 0.f;
    #pragma unroll
    for (int c = 0; c < DIMN; ++c) sdot = fmaf(hi[c], hb[c], sdot);
    float sc = sdot * scale;
    if (sc > m) { l *= __expf(m - sc); m = sc; }
    l += __expf(sc - m);
  }
  sm[threadIdx.x] = m; sl[threadIdx.x] = l;
  __syncthreads();
  for (int s = 128; s > 0; s >>= 1) {
    if (threadIdx.x < s) {
      float ma = sm[threadIdx.x], la = sl[threadIdx.x];
      float mb = sm[threadIdx.x + s], lb = sl[threadIdx.x + s];
      float mm = fmaxf(ma, mb);
      sm[threadIdx.x] = mm;
      sl[threadIdx.x] = la * __expf(ma - mm) + lb * __expf(mb - mm);
    }
    __syncthreads();
  }
  if (threadIdx.x == 0) { mOut[i] = sm[0]; lOut[i] = sl[0]; }
}

// ---------------- attention pass 2: part[i] = (softmax row i) @ hB --------------
__global__ void __launch_bounds__(256)
k_attn_accum(const float* __restrict__ h1, const float* __restrict__ h2,
             const float* __restrict__ m1, const float* __restrict__ l1,
             const float* __restrict__ m2, const float* __restrict__ l2,
             float* __restrict__ part1, float* __restrict__ part2, int n) {
  __shared__ float hi[DIMN];
  __shared__ float red[256 * DIMN];
  const int p = blockIdx.y;
  const float* hA = p ? h2 : h1;
  const float* hB = p ? h1 : h2;
  const float* mIn = p ? m2 : m1;
  const float* lIn = p ? l2 : l1;
  float* part = p ? part2 : part1;
  const int i = blockIdx.x;
  if (threadIdx.x < DIMN) hi[threadIdx.x] = hA[(size_t)i * S + threadIdx.x];
  __syncthreads();
  const float scale = 0.22360679774997896f;
  const float m = mIn[i];
  float acc[DIMN];
  #pragma unroll
  for (int c = 0; c < DIMN; ++c) acc[c] = 0.f;
  for (int j = threadIdx.x; j < n; j += 256) {
    const float* hb = hB + (size_t)j * S;
    float sdot = 0.f;
    #pragma unroll
    for (int c = 0; c < DIMN; ++c) sdot = fmaf(hi[c], hb[c], sdot);
    float pw = __expf(sdot * scale - m);
    #pragma unroll
    for (int c = 0; c < DIMN; ++c) acc[c] = fmaf(pw, hb[c], acc[c]);
  }
  #pragma unroll
  for (int c = 0; c < DIMN; ++c) red[threadIdx.x * DIMN + c] = acc[c];
  __syncthreads();
  for (int s = 128; s > 0; s >>= 1) {
    if (threadIdx.x < s) {
      #pragma unroll
      for (int c = 0; c < DIMN; ++c)
        red[threadIdx.x * DIMN + c] += red[(threadIdx.x + s) * DIMN + c];
    }
    __syncthreads();
  }
  if (threadIdx.x < DIMN)
    part[(size_t)i * DIMN + threadIdx.x] = red[threadIdx.x] * __fdividef(1.f, lIn[i]);
}

// ---------------- mean over rows of part -> om[20] ------------------------------
__global__ void __launch_bounds__(256)
k_mean(const float* __restrict__ part1, const float* __restrict__ part2,
       float* __restrict__ o1m, float* __restrict__ o2m, int n) {
  __shared__ float red[256];
  const int p = blockIdx.y;
  const float* part = p ? part2 : part1;
  float* om = p ? o2m : o1m;
  const int c = blockIdx.x;
  float acc = 0.f;
  for (int i = threadIdx.x; i < n; i += 256) acc += part[(size_t)i * DIMN + c];
  red[threadIdx.x] = acc; __syncthreads();
  for (int s = 128; s > 0; s >>= 1) {
    if (threadIdx.x < s) red[threadIdx.x] += red[threadIdx.x + s];
    __syncthreads();
  }
  if (threadIdx.x == 0) om[c] = red[0] / (float)n;
}

// ---------------- final 2-wide output linear ------------------------------------
__global__ void __launch_bounds__(32)
k_final(const float* __restrict__ o1m, const float* __restrict__ o2m,
        const float* __restrict__ w, const float* __restrict__ b,
        float* __restrict__ out) {
  int c = threadIdx.x;
  if (c >= 2) return;
  float v = b[c];
  for (int d = 0; d < DIMN; ++d) v = fmaf(o1m[d], w[c * 40 + d], v);
  for (int d = 0; d < DIMN; ++d) v = fmaf(o2m[d], w[c * 40 + DIMN + d], v);
  out[c] = v;
}

extern "C" void kernel_launch(void* const* d_in, const int* in_sizes, int n_in,
                              void* d_out, int out_size, void* d_ws, size_t ws_size,
                              hipStream_t stream) {
  const float* adj1   = (const float*)d_in[0];
  const float* adj2   = (const float*)d_in[1];
  const float* esmf1  = (const float*)d_in[2];
  const float* esmf2  = (const float*)d_in[3];
  const float* prot1  = (const float*)d_in[4];
  const float* prot2  = (const float*)d_in[5];
  const float* lin1_w = (const float*)d_in[6];
  const float* lin1_b = (const float*)d_in[7];
  const float* lin2_w = (const float*)d_in[8];
  const float* lin2_b = (const float*)d_in[9];
  const float* wgnn_w = (const float*)d_in[10];
  const float* wgnn_b = (const float*)d_in[11];
  const float* conv1_w = (const float*)d_in[12];
  const float* conv1_b = (const float*)d_in[13];
  const float* conv2_w = (const float*)d_in[14];
  const float* conv2_b = (const float*)d_in[15];
  const float* gru1_wih = (const float*)d_in[16];
  const float* gru1_whh = (const float*)d_in[17];
  const float* gru1_bih = (const float*)d_in[18];
  const float* gru1_bhh = (const float*)d_in[19];
  const float* gru2_wih = (const float*)d_in[20];
  const float* gru2_whh = (const float*)d_in[21];
  const float* gru2_bih = (const float*)d_in[22];
  const float* gru2_bhh = (const float*)d_in[23];
  const float* wesm_w  = (const float*)d_in[24];
  const float* wesm_b  = (const float*)d_in[25];
  const float* wout_w  = (const float*)d_in[26];
  const float* wout_b  = (const float*)d_in[27];
  const float* plam    = (const float*)d_in[28];
  (void)n_in; (void)out_size; (void)ws_size;

  const int n = in_sizes[4] / 7;   // 4096 (N1 == N2)

  float* ws = (float*)d_ws;
  size_t off = 0;
  auto alloc = [&](size_t nf) { float* p = ws + off; off += nf; return p; };
  float* xs1 = alloc((size_t)n * S); float* xs2 = alloc((size_t)n * S);
  float* t1  = alloc((size_t)n * S); float* t2  = alloc((size_t)n * S);
  float* gi1 = alloc((size_t)2 * n * S); float* gi2 = alloc((size_t)2 * n * S);
  float* go1 = alloc((size_t)n * S); float* go2 = alloc((size_t)n * S);
  float* hs1 = alloc((size_t)n * S); float* hs2 = alloc((size_t)n * S);
  float* h1  = alloc((size_t)n * S); float* h2  = alloc((size_t)n * S);
  float* e1  = alloc(S);             float* e2  = alloc(S);
  float* m1  = alloc(n); float* l1 = alloc(n);
  float* m2  = alloc(n); float* l2 = alloc(n);
  float* part1 = alloc((size_t)n * DIMN);
  float* part2 = alloc((size_t)n * DIMN);
  float* o1m = alloc(DIMN); float* o2m = alloc(DIMN);
  float* adjp = alloc((size_t)2 * KS * n * S);   // split-K partials

  const dim3 B(256);
  const int g = (n * S + 255) / 256;

  k_esm<<<dim3(2 * DIMN), B, 0, stream>>>(esmf1, esmf2, wesm_w, wesm_b, e1, e2);
  k_init<<<dim3(g, 2), B, 0, stream>>>(prot1, prot2, lin1_w, lin1_b,
                                       lin2_w, lin2_b, xs1, xs2, n);

  for (int i = 0; i < NL; ++i) {
    k_conv<<<dim3(g, 2), B, 0, stream>>>(xs1, xs2, conv1_w, conv2_w,
                                         conv1_b, conv2_b, t1, t2, n);
    k_gates<<<dim3(2 * g, 2), B, 0, stream>>>(t1, t2, gru1_wih, gru2_wih,
                                              gru1_bih, gru2_bih, gi1, gi2, n);
    k_gru_scan<<<dim3(4), dim3(32), 0, stream>>>(gi1, gi2, gru1_whh, gru2_whh,
                                                 gru1_bhh, gru2_bhh, go1, go2, n);
    k_wgnn<<<dim3(g, 2), B, 0, stream>>>(go1, go2, wgnn_w + i * DIMN * DIMN,
                                         wgnn_b + i * DIMN, hs1, hs2, n);
    k_adj_wmma<<<dim3(n / 128, KS, 2), B, 0, stream>>>(adj1, adj2, hs1, hs2, adjp, n);
    k_adj_red<<<dim3(g, 2), B, 0, stream>>>(adjp, xs1, xs2, n);
  }

  k_fuse<<<dim3(g, 2), B, 0, stream>>>(xs1, xs2, e1, e2, plam, h1, h2, n);
  k_attn_stats<<<dim3(n, 2), B, 0, stream>>>(h1, h2, m1, l1, m2, l2, n);
  k_attn_accum<<<dim3(n, 2), B, 0, stream>>>(h1, h2, m1, l1, m2, l2, part1, part2, n);
  k_mean<<<dim3(DIMN, 2), B, 0, stream>>>(part1, part2, o1m, o2m, n);
  k_final<<<dim3(1), dim3(32), 0, stream>>>(o1m, o2m, wout_w, wout_b, (float*)d_out);
}